// GraphConvolution_49108656062933
// MI455X (gfx1250) — compile-verified
//
#include <hip/hip_runtime.h>

typedef float v2f __attribute__((ext_vector_type(2)));
typedef float v8f __attribute__((ext_vector_type(8)));

#define N_NODES   50000
#define N_EDGES   160000
#define D_FEAT    512
#define N_FILTER  512
#define M_TILES   3125          // 50000 / 16 exactly
#define KCHUNKS   16            // 512 / 32

// ---------------------------------------------------------------------------
// Kernel 1: zero the aggregation workspace (must re-zero every call).
// ---------------------------------------------------------------------------
__global__ void gc_zero_kernel(float4* __restrict__ p, int n4) {
    int i = blockIdx.x * blockDim.x + threadIdx.x;
    const int stride = gridDim.x * blockDim.x;
    const float4 z = make_float4(0.f, 0.f, 0.f, 0.f);
    for (; i < n4; i += stride) p[i] = z;
}

// ---------------------------------------------------------------------------
// Kernel 2: SpMM scatter-add. One block (128 threads) per edge: each thread
// handles one float4 of the 512-float feature row. Hardware f32 atomics at L2.
// ---------------------------------------------------------------------------
__global__ __launch_bounds__(128) void gc_spmm_kernel(
        const float* __restrict__ feat,
        const int*   __restrict__ rows,
        const int*   __restrict__ cols,
        const float* __restrict__ vals,
        float*       __restrict__ agg) {
    const int e = blockIdx.x;
    const int r = rows[e];
    const int c = cols[e];
    const float v = vals[e];

    const int t = threadIdx.x;                 // 0..127, one float4 each
    const float4 f = ((const float4*)(feat + (size_t)c * D_FEAT))[t];
    float* dst = agg + (size_t)r * D_FEAT + 4 * t;
    unsafeAtomicAdd(dst + 0, v * f.x);
    unsafeAtomicAdd(dst + 1, v * f.y);
    unsafeAtomicAdd(dst + 2, v * f.z);
    unsafeAtomicAdd(dst + 3, v * f.w);
}

// ---------------------------------------------------------------------------
// Kernel 3: out = relu(agg @ W + bias) using V_WMMA_F32_16X16X4_F32.
//   block = 128 threads = 4 waves (wave32); wave -> 16(M) x 128(N) strip.
//   B panel DOUBLE-BUFFERED in LDS, pre-paired in fragment layout:
//     lb2[buf][pr][col] = (W[kc+2*pr][col], W[kc+2*pr+1][col])  as float2
//   -> each B fragment is ONE aligned ds_load_b64 into the even VGPR pair.
//   Pipeline: load chunk k+1 W into regs  ->  64 WMMAs on chunk k (hides the
//   global latency)  ->  store regs to other LDS buffer  ->  one barrier.
//   Bank math (stride 144 float2 = 288 dwords): half-waves are one pair-row
//   apart -> bank shift 288 % 64 = 32 -> disjoint bank halves; within a half
//   a b64 spans banks {2c,2c+1}, c=0..15 -> conflict-free fragment loads.
// ---------------------------------------------------------------------------
__global__ __launch_bounds__(128) void gc_gemm_kernel(
        const float* __restrict__ agg,
        const float* __restrict__ W,
        const float* __restrict__ bias,
        float*       __restrict__ out) {
    __shared__ v2f lb2[2][16][144];            // 36.9 KB

    const int tid  = threadIdx.x;
    const int lane = tid & 31;
    const int wave = tid >> 5;
    const int c    = lane & 15;                // column / A-row within tile
    const int hi   = lane >> 4;                // half-wave select

    const int n0    = blockIdx.x * 128;        // N offset of this block
    const int mTile = blockIdx.y * 4 + wave;   // 16-row M tile per wave
    const bool active = (mTile < M_TILES);     // wave-uniform -> EXEC stays full
    const int m0 = mTile * 16;

    // Staging geometry: unit i covers pair-row pr (2 K-rows) x 2 columns.
    const int s_pr0  = tid >> 6;               // pr advances by 2 per unit
    const int s_col  = (tid & 63) * 2;         // even column, fixed per thread
    const float* wBase = W + (size_t)(2 * s_pr0) * N_FILTER + n0 + s_col;

    auto stage_load = [&](int kc, v2f* xr, v2f* yr) {
#pragma unroll
        for (int i = 0; i < 8; ++i) {          // unit i -> pair-row s_pr0 + 2*i
            const float* w0 = wBase + (size_t)(kc + 4 * i) * N_FILTER;
            xr[i] = *(const v2f*)(w0);             // row k,   2 cols
            yr[i] = *(const v2f*)(w0 + N_FILTER);  // row k+1, 2 cols
        }
    };
    auto stage_store = [&](v2f (*buf)[144], const v2f* xr, const v2f* yr) {
#pragma unroll
        for (int i = 0; i < 8; ++i) {
            const int pr = s_pr0 + 2 * i;
            v2f p0; p0.x = xr[i].x; p0.y = yr[i].x;
            v2f p1; p1.x = xr[i].y; p1.y = yr[i].y;
            buf[pr][s_col]     = p0;
            buf[pr][s_col + 1] = p1;
        }
    };

    const v8f vzero = {0.f,0.f,0.f,0.f,0.f,0.f,0.f,0.f};
    v8f acc[8];
#pragma unroll
    for (int j = 0; j < 8; ++j) acc[j] = vzero;

    // A fragment source row for this lane (layout: row = lane&15, K = 2*hi+{0,1})
    const float* aRow = agg + (size_t)(m0 + c) * D_FEAT + 2 * hi;

    v2f xr[8], yr[8];
    stage_load(0, xr, yr);
    stage_store(lb2[0], xr, yr);
    __syncthreads();

    for (int chunk = 0; chunk < KCHUNKS; ++chunk) {
        const int kc = chunk * 32;
        v2f (*cur)[144] = lb2[chunk & 1];
        v2f (*nxt)[144] = lb2[(chunk & 1) ^ 1];
        const bool more = (chunk + 1 < KCHUNKS);

        if (more) stage_load(kc + 32, xr, yr);   // global loads issue early

        if (active) {
#pragma unroll
            for (int ks = 0; ks < 8; ++ks) {
                const v2f a = *(const v2f*)(aRow + kc + 4 * ks); // b64 global
                const int pr = 2 * ks + hi;                      // fragment row
#pragma unroll
                for (int j = 0; j < 8; ++j) {
                    const v2f b = cur[pr][j * 16 + c];           // ds_load_b64
                    acc[j] = __builtin_amdgcn_wmma_f32_16x16x4_f32(
                        /*neg_a=*/false, a, /*neg_b=*/false, b,
                        /*c_mod=*/(short)0, acc[j],
                        /*reuse_a=*/false, /*reuse_b=*/false);
                }
            }
        }

        if (more) stage_store(nxt, xr, yr);      // fill other buffer
        __syncthreads();                          // publish nxt / retire cur
    }

    if (active) {
        // C/D layout: VGPR v -> row m0 + v + 8*hi, column = lane&15 within tile
        const int rbase = m0 + 8 * hi;
#pragma unroll
        for (int j = 0; j < 8; ++j) {
            const int col = n0 + j * 16 + c;
            const float bv = bias[col];
#pragma unroll
            for (int v = 0; v < 8; ++v) {
                float val = acc[j][v] + bv;
                val = val > 0.f ? val : 0.f;
                out[(size_t)(rbase + v) * N_FILTER + col] = val;
            }
        }
    }
}

// ---------------------------------------------------------------------------
extern "C" void kernel_launch(void* const* d_in, const int* in_sizes, int n_in,
                              void* d_out, int out_size, void* d_ws, size_t ws_size,
                              hipStream_t stream) {
    const float* feat = (const float*)d_in[0];
    const int*   erow = (const int*)  d_in[1];
    const int*   ecol = (const int*)  d_in[2];
    const float* eval = (const float*)d_in[3];
    const float* W    = (const float*)d_in[4];
    const float* bias = (const float*)d_in[5];
    float* out = (float*)d_out;
    float* agg = (float*)d_ws;     // N_NODES * D_FEAT floats = 102.4 MB scratch

    const int n4 = (N_NODES * D_FEAT) / 4;
    gc_zero_kernel<<<8192, 256, 0, stream>>>((float4*)agg, n4);

    gc_spmm_kernel<<<N_EDGES, 128, 0, stream>>>(feat, erow, ecol, eval, agg);

    dim3 grid(N_FILTER / 128, (M_TILES + 3) / 4);
    gc_gemm_kernel<<<grid, 128, 0, stream>>>(agg, W, bias, out);
}